// ScaledDotProductAttentionWithBias_75634374082754
// MI455X (gfx1250) — compile-verified
//
#include <hip/hip_runtime.h>

// Scaled dot-product attention with bias+mask, fp32, for MI455X (gfx1250).
// B=2 H=12 S=2048 D=64. Memory-bound: attn write (403MB) + bias read (201MB)
// dominate => ~28us floor at 23.3 TB/s. Matmuls use V_WMMA_F32_16X16X4_F32.

typedef __attribute__((ext_vector_type(2))) float v2f;
typedef __attribute__((ext_vector_type(8))) float v8f;

#define B_ 2
#define H_ 12
#define S_ 2048
#define D_ 64
#define TEMP_INV 0.125f      // 1/TEMPERATURE
#define NEG_FILL -10000.0f
#define SSTRIDE 2052         // 2048 + 4 pad (break 64-bank row aliasing)
#define NWAVES 8
#define NTPW (S_ / 16 / NWAVES)   // 16 score tiles per wave

__global__ __launch_bounds__(256)
void ScaledDotProductAttentionWithBias_75634374082754_kernel(
    const float* __restrict__ q, const float* __restrict__ k,
    const float* __restrict__ v, const int* __restrict__ mask,
    const float* __restrict__ bias, float* __restrict__ out,
    float* __restrict__ attn)
{
  extern __shared__ float smem[];
  float* sc     = smem;                 // [16][SSTRIDE] scores -> probs
  float* red    = smem + 16 * SSTRIDE;  // [16][16] partial reductions
  float* rowmax = red + 256;            // [16]
  float* rowsum = rowmax + 16;          // [16] (holds 1/sum)

  const int tid  = threadIdx.x;
  const int wave = tid >> 5;
  const int lane = tid & 31;
  const int l16  = lane & 15;
  const int hi   = lane >> 4;   // 0: lanes 0-15, 1: lanes 16-31
  const int c0   = hi << 1;     // K sub-offset for A/B frags: 0 or 2

  const int blk   = blockIdx.x;
  const int qt    = blk & 127;  // 128 q-tiles per (b,h)
  const int bh    = blk >> 7;   // 0..23
  const int b     = bh / H_;
  const int h     = bh % H_;
  const int qbase = qt * 16;

  const float* qp = q    + (size_t)bh * S_ * D_;
  const float* kp = k    + (size_t)bh * S_ * D_;
  const float* vp = v    + (size_t)bh * S_ * D_;
  const float* bp = bias + (size_t)h * S_ * S_ + (size_t)qbase * S_;
  const int*   mp = mask + b * S_;
  float* outp = out  + (size_t)bh * S_ * D_ + (size_t)qbase * D_;
  float* attp = attn + (size_t)bh * S_ * S_ + (size_t)qbase * S_;

  // ---- Preload Q A-fragments (16x4 f32 per K-step), fold in 1/T ----
  // A layout: VGPR j, lane L: M = L%16, K = 4*s + (L<16 ? j : j+2)
  v2f aq[16];
  {
    const float* qr = qp + (size_t)(qbase + l16) * D_;
    #pragma unroll
    for (int s = 0; s < 16; ++s) {
      v2f t = *(const v2f*)(qr + s * 4 + c0);   // 8B aligned
      aq[s].x = t.x * TEMP_INV;
      aq[s].y = t.y * TEMP_INV;
    }
  }

  // ---- Phase 1: scores = (Q/T)K^T + bias, masked -> LDS ----
  for (int i = 0; i < NTPW; ++i) {
    const int nbase = (wave * NTPW + i) * 16;
    const int col   = nbase + l16;
    const float* kr = kp + (size_t)col * D_;
    v8f acc = {};
    #pragma unroll
    for (int s = 0; s < 16; ++s) {            // D=64 in steps of 4
      // B layout: VGPR j, lane L: K = 4*s + (L<16 ? j : j+2), N = L%16
      v2f bf = *(const v2f*)(kr + s * 4 + c0);
      acc = __builtin_amdgcn_wmma_f32_16x16x4_f32(
          false, aq[s], false, bf, (short)0, acc, false, false);
    }
    const int mk = mp[col];
    #pragma unroll
    for (int r = 0; r < 8; ++r) {             // C: VGPR r -> M = r + 8*hi
      const int m = r + hi * 8;
      float sv = acc[r] + bp[(size_t)m * S_ + col];
      if (mk == 0) sv = NEG_FILL;
      sc[m * SSTRIDE + col] = sv;
    }
  }
  __syncthreads();

  // ---- Phase 2: softmax over each 2048-wide row ----
  {
    const int row = tid >> 4, seg = tid & 15;
    const float* p = sc + row * SSTRIDE + seg * 128;
    float mx = -3.4e38f;
    for (int j = 0; j < 128; ++j) mx = fmaxf(mx, p[j]);
    red[row * 16 + seg] = mx;
  }
  __syncthreads();
  if (tid < 16) {
    float mx = red[tid * 16];
    for (int j = 1; j < 16; ++j) mx = fmaxf(mx, red[tid * 16 + j]);
    rowmax[tid] = mx;
  }
  __syncthreads();
  {
    const int row = tid >> 4, seg = tid & 15;
    float* p = sc + row * SSTRIDE + seg * 128;
    const float mx = rowmax[row];
    float sum = 0.f;
    for (int j = 0; j < 128; ++j) {
      float e = __expf(p[j] - mx);
      p[j] = e;
      sum += e;
    }
    red[row * 16 + seg] = sum;
  }
  __syncthreads();
  if (tid < 16) {
    float sm = 0.f;
    for (int j = 0; j < 16; ++j) sm += red[tid * 16 + j];
    rowsum[tid] = 1.0f / sm;
  }
  __syncthreads();

  // ---- Phase 3: normalize in LDS + coalesced attn store (dominant stream) ----
  for (int row = 0; row < 16; ++row) {
    const float inv = rowsum[row];
    float* prow = sc + row * SSTRIDE;
    float* arow = attp + (size_t)row * S_;
    for (int c = tid; c < S_; c += 256) {   // 256 lanes = 1KB contiguous
      float pv = prow[c] * inv;
      prow[c] = pv;
      arow[c] = pv;
    }
  }
  __syncthreads();

  // ---- Phase 4: out = P @ V, K split across waves ----
  v8f oacc[4] = {};
  {
    const int kbase = wave * (S_ / NWAVES);      // 256 K per wave
    for (int ks = 0; ks < S_ / NWAVES; ks += 4) {
      const int kk = kbase + ks;
      v2f af = *(const v2f*)(sc + l16 * SSTRIDE + kk + c0);  // ds_load_b64
      #pragma unroll
      for (int n = 0; n < 4; ++n) {              // N = 64 in 4 tiles
        v2f bf;
        bf.x = vp[(size_t)(kk + c0)     * D_ + n * 16 + l16];
        bf.y = vp[(size_t)(kk + c0 + 1) * D_ + n * 16 + l16];
        oacc[n] = __builtin_amdgcn_wmma_f32_16x16x4_f32(
            false, af, false, bf, (short)0, oacc[n], false, false);
      }
    }
  }
  __syncthreads();  // all waves done reading P; safe to reuse sc
  #pragma unroll
  for (int n = 0; n < 4; ++n) {
    #pragma unroll
    for (int r = 0; r < 8; ++r) {
      const int m = r + hi * 8;
      sc[wave * 1024 + m * 64 + n * 16 + l16] = oacc[n][r];
    }
  }
  __syncthreads();
  for (int e = tid; e < 1024; e += 256) {        // reduce 8 partials
    float s = 0.f;
    #pragma unroll
    for (int w = 0; w < 8; ++w) s += sc[w * 1024 + e];
    outp[e] = s;                                  // D_=64 so flat index works
  }
}

extern "C" void kernel_launch(void* const* d_in, const int* in_sizes, int n_in,
                              void* d_out, int out_size, void* d_ws, size_t ws_size,
                              hipStream_t stream) {
  (void)in_sizes; (void)n_in; (void)out_size; (void)d_ws; (void)ws_size;
  const float* q    = (const float*)d_in[0];
  const float* k    = (const float*)d_in[1];
  const float* v    = (const float*)d_in[2];
  const int*   mask = (const int*)d_in[3];
  const float* bias = (const float*)d_in[4];
  float* out  = (float*)d_out;
  float* attn = out + (size_t)B_ * H_ * S_ * D_;   // tuple order: (out, attn)

  const size_t shmem = (size_t)(16 * SSTRIDE + 256 + 32) * sizeof(float); // ~129.4KB
  // Opt in to >64KB dynamic LDS (gfx1250 WGP has 320KB). Deterministic per call.
  (void)hipFuncSetAttribute(
      (const void*)ScaledDotProductAttentionWithBias_75634374082754_kernel,
      hipFuncAttributeMaxDynamicSharedMemorySize, (int)shmem);

  const int nblocks = B_ * H_ * (S_ / 16);  // 3072
  ScaledDotProductAttentionWithBias_75634374082754_kernel
      <<<nblocks, 256, shmem, stream>>>(q, k, v, mask, bias, out, attn);
}